// NTMMemory_65051574665615
// MI455X (gfx1250) — compile-verified
//
#include <hip/hip_runtime.h>
#include <math.h>

typedef __attribute__((ext_vector_type(2))) float v2f;
typedef __attribute__((ext_vector_type(4))) float v4f;
typedef __attribute__((ext_vector_type(8))) float v8f;

#define B_ 256
#define N_ 4096
#define M_ 64
#define NTHREADS 256
#define NWAVES 8
#define TILES_PER_WAVE (N_ / 16 / NWAVES)   // 32 tiles of 16 rows per wave

__global__ __launch_bounds__(NTHREADS)
void ntm_memory_kernel(const float* __restrict__ memory,   // [B,N,M]
                       const float* __restrict__ kvec,     // [B,M]
                       const float* __restrict__ beta_p,   // [B,1]
                       const float* __restrict__ g_p,      // [B,1]
                       const float* __restrict__ s_p,      // [B,3]
                       const float* __restrict__ gamma_p,  // [B,1]
                       const float* __restrict__ w_prev,   // [B,N]
                       const float* __restrict__ e_p,      // [B,M]
                       const float* __restrict__ a_p,      // [B,M]
                       float* __restrict__ out_rd,         // [B,M]
                       float* __restrict__ out_mem,        // [B,N,M]
                       float* __restrict__ out_w)          // [B,N]
{
    __shared__ float k_s[M_];
    __shared__ float wln[N_];      // logits -> exp -> wg (in place, barrier-separated)
    __shared__ float wps[N_];      // sharpened -> normalized weights
    __shared__ float red[NTHREADS];
    __shared__ v4f   red4[NTHREADS];   // phase-B partial rd vectors

    const int tid  = threadIdx.x;
    const int b    = blockIdx.x;
    const int lane = tid & 31;
    const int wv   = tid >> 5;
    const int half = lane >> 4;    // 0: lanes 0-15, 1: lanes 16-31
    const int row  = lane & 15;    // A-matrix row owned by this lane
    const float EPS = 1e-16f;

    // per-batch scalars
    const float beta  = beta_p[b];
    const float g     = g_p[b];
    const float s0    = s_p[b * 3 + 0];
    const float s1    = s_p[b * 3 + 1];
    const float s2    = s_p[b * 3 + 2];
    const float gamma = gamma_p[b];

    if (tid < M_) k_s[tid] = kvec[(size_t)b * M_ + tid] + EPS;   // k_e = k + eps
    __syncthreads();

    // ||k_e|| (cheap, computed redundantly per thread)
    float kn2 = 0.f;
    #pragma unroll
    for (int i = 0; i < M_; ++i) { float kv = k_s[i]; kn2 = fmaf(kv, kv, kn2); }
    const float kn = fmaxf(sqrtf(kn2), 1e-8f);

    // Preload B fragments: k replicated across all 16 columns.
    // f32 4x16 B layout mirrors A: VGPR v at lane L holds K = 4c + v + 2*(L>>4).
    v2f bfrag[16];
    #pragma unroll
    for (int c = 0; c < 16; ++c) {
        bfrag[c].x = k_s[4 * c + 0 + 2 * half];
        bfrag[c].y = k_s[4 * c + 1 + 2 * half];
    }

    const float* memb = memory + (size_t)b * N_ * M_;

    // ---------------- Phase A: content logits via V_WMMA_F32_16X16X4_F32 ----
    for (int t = 0; t < TILES_PER_WAVE; ++t) {
        const int n0 = (wv * TILES_PER_WAVE + t) * 16;
        const float* rowp = memb + (size_t)(n0 + row) * M_;

        v8f acc = {0.f, 0.f, 0.f, 0.f, 0.f, 0.f, 0.f, 0.f};
        float nrm = 0.f;
        #pragma unroll
        for (int c = 0; c < 16; ++c) {
            // A fragment: lane L -> mem_e[row, 4c + {0,1} + 2*half]
            const float* p = rowp + 4 * c + 2 * half;
            v2f av;
            av.x = p[0] + EPS;
            av.y = p[1] + EPS;
            nrm = fmaf(av.x, av.x, nrm);
            nrm = fmaf(av.y, av.y, nrm);
            // D = A(16x4) * B(4x16) + C  ; all 16 D columns equal num[row]
            acc = __builtin_amdgcn_wmma_f32_16x16x4_f32(
                false, av, false, bfrag[c], (short)0, acc, false, false);
        }
        // full row norm: lanes L and L^16 jointly hold the 64 elements of row L&15
        nrm += __shfl_xor(nrm, 16, 32);

        // D layout: lane L, VGPR v -> (M = v + 8*half, N = L%16).
        // Column-0 lanes (0 and 16) publish the 16 logits of this tile.
        #pragma unroll
        for (int v = 0; v < 8; ++v) {
            float nv = __shfl(nrm, 8 * half + v, 32);   // norm of row v+8*half
            if (row == 0) {
                float mem_n = fmaxf(sqrtf(nv), 1e-8f);
                wln[n0 + v + 8 * half] = beta * (acc[v] / (mem_n * kn));
            }
        }
    }
    __syncthreads();

    // ---------------- softmax over N (stable) -------------------------------
    float lmax = -INFINITY;
    #pragma unroll
    for (int j = 0; j < N_ / NTHREADS; ++j)
        lmax = fmaxf(lmax, wln[tid + j * NTHREADS]);
    red[tid] = lmax; __syncthreads();
    for (int off = NTHREADS / 2; off > 0; off >>= 1) {
        if (tid < off) red[tid] = fmaxf(red[tid], red[tid + off]);
        __syncthreads();
    }
    const float mx = red[0]; __syncthreads();

    float lsum = 0.f;
    #pragma unroll
    for (int j = 0; j < N_ / NTHREADS; ++j) {
        int n = tid + j * NTHREADS;
        float ev = __expf(wln[n] - mx);
        wln[n] = ev;
        lsum += ev;
    }
    red[tid] = lsum; __syncthreads();
    for (int off = NTHREADS / 2; off > 0; off >>= 1) {
        if (tid < off) red[tid] += red[tid + off];
        __syncthreads();
    }
    const float denom = red[0]; __syncthreads();

    // wg = g*wc + (1-g)*w_prev   (in place in wln)
    #pragma unroll
    for (int j = 0; j < N_ / NTHREADS; ++j) {
        int n = tid + j * NTHREADS;
        float wc = wln[n] / denom;
        wln[n] = g * wc + (1.f - g) * w_prev[(size_t)b * N_ + n];
    }
    __syncthreads();

    // circular 3-tap conv + sharpen
    float psum = 0.f;
    #pragma unroll
    for (int j = 0; j < N_ / NTHREADS; ++j) {
        int n  = tid + j * NTHREADS;
        int nm = (n == 0) ? (N_ - 1) : (n - 1);
        int np = (n == N_ - 1) ? 0 : (n + 1);
        float wh = s0 * wln[nm] + s1 * wln[n] + s2 * wln[np];
        float wp = powf(wh, gamma);
        wps[n] = wp;
        psum  += wp;
    }
    red[tid] = psum; __syncthreads();
    for (int off = NTHREADS / 2; off > 0; off >>= 1) {
        if (tid < off) red[tid] += red[tid + off];
        __syncthreads();
    }
    const float wsum = red[0] + EPS; __syncthreads();

    #pragma unroll
    for (int j = 0; j < N_ / NTHREADS; ++j) {
        int n = tid + j * NTHREADS;
        float wn = wps[n] / wsum;
        wps[n] = wn;
        out_w[(size_t)b * N_ + n] = wn;
    }
    __syncthreads();

    // ---------------- Phase B: fused read + erase/add rewrite ---------------
    // thread -> (m4 = (tid&15)*4, n-group = tid>>4): 16B/lane, fully coalesced
    // 256B row segments per half-wave. Last use of `memory`, write-once
    // `new_mem` -> stream both with non-temporal hints to spare L2.
    const int m4  = (tid & 15) * 4;
    const int grp = tid >> 4;                        // 0..15
    const v4f ev4 = *(const v4f*)(e_p + (size_t)b * M_ + m4);
    const v4f av4 = *(const v4f*)(a_p + (size_t)b * M_ + m4);
    v4f rdacc = {0.f, 0.f, 0.f, 0.f};
    float* omem = out_mem + (size_t)b * N_ * M_;
    for (int n = grp; n < N_; n += 16) {
        const float wn = wps[n];
        const v4f mval = __builtin_nontemporal_load(
            (const v4f*)(memb + (size_t)n * M_ + m4));
        rdacc += wn * mval;
        v4f nv = mval * (1.f - wn * ev4) + wn * av4;
        __builtin_nontemporal_store(nv, (v4f*)(omem + (size_t)n * M_ + m4));
    }
    red4[tid] = rdacc; __syncthreads();
    if (tid < M_) {
        // rd[m]: contributions live at threads {grp*16 + (m>>2)}, component m&3
        const int col = tid >> 2, cmp = tid & 3;
        float acc = 0.f;
        #pragma unroll
        for (int gi = 0; gi < 16; ++gi)
            acc += red4[gi * 16 + col][cmp];
        out_rd[(size_t)b * M_ + tid] = acc;
    }
}

extern "C" void kernel_launch(void* const* d_in, const int* in_sizes, int n_in,
                              void* d_out, int out_size, void* d_ws, size_t ws_size,
                              hipStream_t stream) {
    (void)in_sizes; (void)n_in; (void)d_ws; (void)ws_size; (void)out_size;
    const float* memory = (const float*)d_in[0];
    const float* kvec   = (const float*)d_in[1];
    const float* beta   = (const float*)d_in[2];
    const float* g      = (const float*)d_in[3];
    const float* s      = (const float*)d_in[4];
    const float* gamma  = (const float*)d_in[5];
    const float* w_prev = (const float*)d_in[6];
    const float* e      = (const float*)d_in[7];
    const float* a      = (const float*)d_in[8];

    // outputs concatenated flat in return order: rd [B,M], new_mem [B,N,M], w [B,N]
    float* out_rd  = (float*)d_out;
    float* out_mem = out_rd + (size_t)B_ * M_;
    float* out_w   = out_mem + (size_t)B_ * N_ * M_;

    ntm_memory_kernel<<<B_, NTHREADS, 0, stream>>>(
        memory, kvec, beta, g, s, gamma, w_prev, e, a, out_rd, out_mem, out_w);
}